// GNN_59459527246386
// MI455X (gfx1250) — compile-verified
//
#include <hip/hip_runtime.h>
#include <stdint.h>

typedef __attribute__((ext_vector_type(2))) float v2f;
typedef __attribute__((ext_vector_type(8))) float v8f;

#define N_NODES  50000
#define N_EDGES  800000
#define FDIM     128
#define N_GRAPHS 512
#define N_STRIPS (N_NODES / 16)   // 3125 exactly
#define KPAIRS   (FDIM / 2)       // 64 K-pairs

// ---------------------------------------------------------------- utilities
__global__ void fill_f32(float* __restrict__ p, float v, int n) {
    int i = blockIdx.x * blockDim.x + threadIdx.x;
    if (i < n) p[i] = v;
}

__global__ void deg_kernel(const long long* __restrict__ dst, float* __restrict__ deg) {
    int e = blockIdx.x * blockDim.x + threadIdx.x;
    if (e < N_EDGES) atomicAdd(&deg[dst[e]], 1.0f);
}

__global__ void dinv_kernel(float* __restrict__ deg) {
    int i = blockIdx.x * blockDim.x + threadIdx.x;
    if (i < N_NODES) deg[i] = rsqrtf(fmaxf(deg[i], 1e-12f));
}

// ------------------------------------------------------- WMMA f32 GEMM (x@W)
// A: [N_NODES, 128] row-major, W: [128, 128] row-major, H: [N_NODES, 128].
// Block = 256 threads = 8 wave32. W is staged into LDS in a fragment-native
// packed layout: Wp[p][n] = {W[2p][n], W[2p+1][n]} so every WMMA B-fragment
// is a single 8B-aligned ds_load_b64 straight into the operand pair.
// Each wave computes one 16-row strip across all 128 output columns.
template <bool RELU_IN>
__global__ __launch_bounds__(256) void gemm_wmma(const float* __restrict__ A,
                                                 const float* __restrict__ W,
                                                 float* __restrict__ H) {
    __shared__ v2f Wp[KPAIRS * FDIM];          // 64 KB of the 320 KB WGP LDS
    // cooperative packed stage: idx = p*128 + n; threads consecutive in n -> coalesced
    for (int idx = threadIdx.x; idx < KPAIRS * FDIM; idx += 256) {
        int p = idx >> 7, n = idx & 127;
        v2f w;
        w.x = W[(2 * p) * FDIM + n];
        w.y = W[(2 * p + 1) * FDIM + n];
        Wp[idx] = w;
    }
    __syncthreads();

    const int wave  = threadIdx.x >> 5;        // wave32
    const int lane  = threadIdx.x & 31;
    const int strip = blockIdx.x * 8 + wave;
    if (strip >= N_STRIPS) return;             // uniform per wave -> EXEC all-1s inside

    const int m0   = strip * 16;
    const int half = lane >> 4;                // 0: K={0,1}, 1: K={2,3} (ISA 16x4 f32 A layout)
    const int l    = lane & 15;

    v8f acc[8];
#pragma unroll
    for (int t = 0; t < 8; ++t) acc[t] = {};

    const float* arow = A + (size_t)(m0 + l) * FDIM;

#pragma unroll 4
    for (int k0 = 0; k0 < FDIM; k0 += 4) {
        const int kk = k0 + 2 * half;          // lane's K pair start
        const int pi = (k0 >> 1) + half;       // packed pair index
        v2f a = *(const v2f*)(arow + kk);      // single global_load_b64
        if (RELU_IN) { a.x = fmaxf(a.x, 0.0f); a.y = fmaxf(a.y, 0.0f); }
        const v2f* brow = &Wp[pi * FDIM + l];  // tiles at +16 elem = +128B imm offsets
#pragma unroll
        for (int t = 0; t < 8; ++t) {
            v2f b = brow[t * 16];              // single ds_load_b64, contiguous pair
            acc[t] = __builtin_amdgcn_wmma_f32_16x16x4_f32(
                false, a, false, b, (short)0, acc[t], false, false);
        }
    }

    // D layout: VGPR r -> row m0+r (lanes 0-15) / m0+8+r (lanes 16-31), col = n0+l
#pragma unroll
    for (int t = 0; t < 8; ++t) {
        const int n0 = t * 16;
#pragma unroll
        for (int r = 0; r < 8; ++r) {
            H[(size_t)(m0 + r + 8 * half) * FDIM + n0 + l] = acc[t][r];
        }
    }
}

// --------------------------------------------- aggregation: self-loop + bias
__global__ void selfloop_bias(const float* __restrict__ h, const float* __restrict__ dinv,
                              const float* __restrict__ b, float* __restrict__ out) {
    int i = blockIdx.x * blockDim.x + threadIdx.x;
    if (i < N_NODES * FDIM) {
        int node = i >> 7, f = i & 127;
        float di = dinv[node];
        out[i] = h[i] * di * di + b[f];
    }
}

// ---------------------------------------- aggregation: edge gather + scatter
// One wave32 per edge; each lane moves 4 consecutive floats (float4 gather,
// 4x f32 global atomics).
__global__ __launch_bounds__(256) void edge_agg(const float* __restrict__ h,
                                                const long long* __restrict__ src,
                                                const long long* __restrict__ dst,
                                                const float* __restrict__ dinv,
                                                float* __restrict__ out) {
    int wave = (int)((blockIdx.x * (unsigned)blockDim.x + threadIdx.x) >> 5);
    int lane = threadIdx.x & 31;
    if (wave >= N_EDGES) return;
    long long s = src[wave], d = dst[wave];
    float norm = dinv[s] * dinv[d];
    __builtin_prefetch(out + (size_t)d * FDIM, 0, 1);   // global_prefetch_b8
    float4 v = ((const float4*)(h + (size_t)s * FDIM))[lane];
    float* o = out + (size_t)d * FDIM + lane * 4;
    atomicAdd(o + 0, v.x * norm);
    atomicAdd(o + 1, v.y * norm);
    atomicAdd(o + 2, v.z * norm);
    atomicAdd(o + 3, v.w * norm);
}

// ------------------------------------------------------ pooling + final head
__global__ void pool_kernel(const float* __restrict__ z, const long long* __restrict__ batch,
                            float* __restrict__ sums, float* __restrict__ cnts) {
    int node = blockIdx.x;
    int t    = threadIdx.x;
    long long g = batch[node];
    float v = fmaxf(z[(size_t)node * FDIM + t], 0.0f);  // final ReLU fused here
    atomicAdd(&sums[(size_t)g * FDIM + t], v);
    if (t == 0) atomicAdd(&cnts[g], 1.0f);
}

__global__ void final_lin(const float* __restrict__ sums, const float* __restrict__ cnts,
                          const float* __restrict__ Wlin, const float* __restrict__ blin,
                          float* __restrict__ out) {
    __shared__ float r0[FDIM];
    __shared__ float r1[FDIM];
    int g = blockIdx.x, t = threadIdx.x;
    float p = sums[(size_t)g * FDIM + t] / fmaxf(cnts[g], 1.0f);
    r0[t] = p * Wlin[t * 2 + 0];
    r1[t] = p * Wlin[t * 2 + 1];
    __syncthreads();
    for (int s = 64; s > 0; s >>= 1) {
        if (t < s) { r0[t] += r0[t + s]; r1[t] += r1[t + s]; }
        __syncthreads();
    }
    if (t == 0) {
        out[g * 2 + 0] = r0[0] + blin[0];
        out[g * 2 + 1] = r1[0] + blin[1];
    }
}

// ---------------------------------------------------------------- dispatch
extern "C" void kernel_launch(void* const* d_in, const int* in_sizes, int n_in,
                              void* d_out, int out_size, void* d_ws, size_t ws_size,
                              hipStream_t stream) {
    (void)in_sizes; (void)n_in; (void)out_size; (void)ws_size;
    const float*     x     = (const float*)d_in[0];
    const long long* ei    = (const long long*)d_in[1];   // int64 [2, N_EDGES]
    const long long* batch = (const long long*)d_in[2];   // int64 [N_NODES]
    const float*     W1    = (const float*)d_in[3];
    const float*     b1    = (const float*)d_in[4];
    const float*     W2    = (const float*)d_in[5];
    const float*     b2    = (const float*)d_in[6];
    const float*     Wlin  = (const float*)d_in[7];
    const float*     blin  = (const float*)d_in[8];
    float*           out   = (float*)d_out;

    const long long* src = ei;
    const long long* dst = ei + N_EDGES;

    // carve workspace (256B-aligned slabs)
    char*  ws  = (char*)d_ws;
    size_t off = 0;
    auto alloc = [&](size_t bytes) {
        void* p = ws + off;
        off = (off + bytes + 255) & ~(size_t)255;
        return p;
    };
    float* dinv = (float*)alloc((size_t)N_NODES * 4);
    float* bufA = (float*)alloc((size_t)N_NODES * FDIM * 4);  // GEMM outputs h
    float* bufB = (float*)alloc((size_t)N_NODES * FDIM * 4);  // aggregated z
    float* sums = (float*)alloc((size_t)N_GRAPHS * FDIM * 4);
    float* cnts = (float*)alloc((size_t)N_GRAPHS * 4);

    // degrees -> dinv (deg starts at 1.0 for the self-loop)
    fill_f32<<<(N_NODES + 255) / 256, 256, 0, stream>>>(dinv, 1.0f, N_NODES);
    deg_kernel<<<(N_EDGES + 255) / 256, 256, 0, stream>>>(dst, dinv);
    dinv_kernel<<<(N_NODES + 255) / 256, 256, 0, stream>>>(dinv);

    const int gemm_blocks = (N_STRIPS + 7) / 8;   // 391
    const int edge_blocks = (N_EDGES + 7) / 8;    // 8 waves (edges) per block

    // ---- layer 1: h1 = x @ W1 ; z1 = scatter(h1) + b1
    gemm_wmma<false><<<gemm_blocks, 256, 0, stream>>>(x, W1, bufA);
    selfloop_bias<<<(N_NODES * FDIM + 255) / 256, 256, 0, stream>>>(bufA, dinv, b1, bufB);
    edge_agg<<<edge_blocks, 256, 0, stream>>>(bufA, src, dst, dinv, bufB);

    // ---- layer 2: h2 = relu(z1) @ W2 ; z2 = scatter(h2) + b2
    gemm_wmma<true><<<gemm_blocks, 256, 0, stream>>>(bufB, W2, bufA);
    selfloop_bias<<<(N_NODES * FDIM + 255) / 256, 256, 0, stream>>>(bufA, dinv, b2, bufB);
    edge_agg<<<edge_blocks, 256, 0, stream>>>(bufA, src, dst, dinv, bufB);

    // ---- pool (mean over graphs, relu fused) + linear head
    fill_f32<<<(N_GRAPHS * FDIM + 255) / 256, 256, 0, stream>>>(sums, 0.0f, N_GRAPHS * FDIM);
    fill_f32<<<(N_GRAPHS + 255) / 256, 256, 0, stream>>>(cnts, 0.0f, N_GRAPHS);
    pool_kernel<<<N_NODES, FDIM, 0, stream>>>(bufB, batch, sums, cnts);
    final_lin<<<N_GRAPHS, FDIM, 0, stream>>>(sums, cnts, Wlin, blin, out);
}